// CgpHmmLayer_16784732193333
// MI455X (gfx1250) — compile-verified
//
#include <hip/hip_runtime.h>
#include <math.h>

// ---------------- problem constants (from reference) ----------------
#define N_STATES 132     // 1+3+60+3+1+63+1
#define ALPH     126
#define BATCH    32
#define TLEN     10000
#define KSTEPS   33      // 132 / 4  (V_WMMA_F32_16X16X4_F32 -> K=4, exact fit)
#define NTILES   9       // ceil(132/16) -> N padded to 144
#define ROWS_PER_WG 16   // batch rows per workgroup (one 16xN WMMA M-tile)
#define ASTRIDE  148     // alpha row stride (floats): bank-conflict-free, 16B-mult

typedef __attribute__((ext_vector_type(2))) float v2f;
typedef __attribute__((ext_vector_type(8))) float v8f;

// ---------------- kernel 1: one-hot -> index decode ----------------
// one wave per (b,t); the 161MB one-hot tensor is read exactly once.
__global__ void decode_onehot_kernel(const float* __restrict__ oh,
                                     int* __restrict__ obs, int total) {
  int gw   = (blockIdx.x * blockDim.x + threadIdx.x) >> 5;
  int lane = threadIdx.x & 31;
  if (gw >= total) return;
  const float* p = oh + (size_t)gw * ALPH;
  int cand = 0;
  #pragma unroll
  for (int r = 0; r < 4; ++r) {
    int j = lane + 32 * r;
    if (j < ALPH && p[j] > 0.5f) cand = j;
  }
  #pragma unroll
  for (int off = 16; off > 0; off >>= 1)
    cand = max(cand, __shfl_down(cand, off, 32));
  if (lane == 0) obs[gw] = cand;
}

// ---------------- kernel 2: scaled HMM forward recursion ----------------
// 2 blocks x 288 threads (9 waves). Wave w owns state columns [16w,16w+16).
// Transition matrix fragments live in registers for all 10000 steps.
__global__ __launch_bounds__(288)
void hmm_forward_kernel(const float* __restrict__ Ag,   // [132][132] row-major
                        const float* __restrict__ Bg,   // [132][126]
                        const float* __restrict__ I0,   // [132]
                        const int*   __restrict__ obsw, // [32][10000]
                        float* __restrict__ out) {      // [32]
  __shared__ __align__(16) float abuf[2][ROWS_PER_WG * ASTRIDE]; // double-buffered alpha (raw)
  __shared__ float cbuf[ROWS_PER_WG];    // 1/c per row, applied at next step's reads
  __shared__ int   obs_sh[ROWS_PER_WG];  // obs symbol for the step being computed

  const int tid = threadIdx.x;
  const int ln  = tid & 15;              // lane % 16
  const int hi  = (tid >> 4) & 1;        // lane half (K/M split per WMMA layout)
  const int wv  = tid >> 5;              // wave id 0..8
  const int n0  = wv * 16;
  const int n   = n0 + ln;               // this lane's state column
  const bool nvalid = (n < N_STATES);
  const int gb0 = blockIdx.x * ROWS_PER_WG;

  // Preload B-operand (transition matrix) fragments: 33 k-steps x v2f = 66 VGPRs.
  // Layout: vgpr0 = A[k0+2*hi][n], vgpr1 = A[k0+2*hi+1][n]; zero for padded cols.
  v2f bfrag[KSTEPS];
  #pragma unroll
  for (int kk = 0; kk < KSTEPS; ++kk) {
    int k = 4 * kk + 2 * hi;
    bfrag[kk].x = nvalid ? Ag[k * N_STATES + n]       : 0.0f;
    bfrag[kk].y = nvalid ? Ag[(k + 1) * N_STATES + n] : 0.0f;
  }

  // ---- t = 0: alpha0 = I0[s] * B[s, obs0] ----
  for (int i = tid; i < ROWS_PER_WG * N_STATES; i += (int)blockDim.x) {
    int m = i / N_STATES, s = i - m * N_STATES;
    int o = obsw[(size_t)(gb0 + m) * TLEN];
    abuf[0][m * ASTRIDE + s] = I0[s] * Bg[s * ALPH + o];
  }
  __syncthreads();
  float ll = 0.0f;
  if (tid < ROWS_PER_WG) {
    const float4* rp = (const float4*)&abuf[0][tid * ASTRIDE];
    float s = 0.0f;
    for (int i = 0; i < KSTEPS; ++i) { float4 v = rp[i]; s += v.x + v.y + v.z + v.w; }
    cbuf[tid] = 1.0f / s;
    ll = logf(s);
    obs_sh[tid] = obsw[(size_t)(gb0 + tid) * TLEN + 1];
  }
  __syncthreads();

  int cur = 0;
  for (int t = 1; t < TLEN; ++t) {
    const float icm = cbuf[ln];          // normalization folded into A-fragment reads
    const float* ac = abuf[cur];
    float* an = abuf[cur ^ 1];

    v8f acc0 = {}; v8f acc1 = {};        // split accumulators: halve dependent WMMA chain
    #pragma unroll
    for (int kk = 0; kk < KSTEPS; ++kk) {
      int k = 4 * kk + 2 * hi;
      v2f a;                             // A-operand: alpha[m=ln][k..k+1] (ds_load_b64)
      a.x = ac[ln * ASTRIDE + k]     * icm;
      a.y = ac[ln * ASTRIDE + k + 1] * icm;
      if (kk & 1)
        acc1 = __builtin_amdgcn_wmma_f32_16x16x4_f32(false, a, false, bfrag[kk],
                                                     (short)0, acc1, false, false);
      else
        acc0 = __builtin_amdgcn_wmma_f32_16x16x4_f32(false, a, false, bfrag[kk],
                                                     (short)0, acc0, false, false);
    }

    // Emission gather (B hot in WGP$) + raw alpha write; padded cols -> 0.
    #pragma unroll
    for (int r = 0; r < 8; ++r) {
      int m = r + 8 * hi;                // C/D layout: vgpr r, lanes16-31 -> M=r+8
      int o = obs_sh[m];
      float e = nvalid ? (acc0[r] + acc1[r]) * Bg[n * ALPH + o] : 0.0f;
      an[m * ASTRIDE + n] = e;
    }
    __syncthreads();                     // raw alpha visible; all reads of prev state done

    if (tid < ROWS_PER_WG) {
      const float4* rp = (const float4*)&an[tid * ASTRIDE];
      float s = 0.0f;
      for (int i = 0; i < KSTEPS; ++i) { float4 v = rp[i]; s += v.x + v.y + v.z + v.w; }
      cbuf[tid] = 1.0f / s;
      ll += logf(s);
      if (t + 1 < TLEN) obs_sh[tid] = obsw[(size_t)(gb0 + tid) * TLEN + t + 1];
    }
    __syncthreads();                     // cbuf / obs_sh visible for next step
    cur ^= 1;
  }

  if (tid < ROWS_PER_WG) out[gb0 + tid] = ll;
}

// ---------------- host-side launch ----------------
extern "C" void kernel_launch(void* const* d_in, const int* in_sizes, int n_in,
                              void* d_out, int out_size, void* d_ws, size_t ws_size,
                              hipStream_t stream) {
  const float* oh = (const float*)d_in[0];  // [32,10000,126] one-hot
  const float* A  = (const float*)d_in[1];  // [132,132]
  const float* B  = (const float*)d_in[2];  // [132,126]
  const float* I0 = (const float*)d_in[3];  // [132]
  float* out = (float*)d_out;               // [32]
  int* obs = (int*)d_ws;                    // [32*10000] decoded symbols

  const int total = BATCH * TLEN;           // 320000 (b,t) pairs, one wave each
  const int wavesPerBlock = 8;              // 256 threads
  const int blocks = (total + wavesPerBlock - 1) / wavesPerBlock;
  decode_onehot_kernel<<<blocks, 256, 0, stream>>>(oh, obs, total);

  hmm_forward_kernel<<<2, 288, 0, stream>>>(A, B, I0, obs, out);
}